// MPS_RNN_2D_86088324481212
// MI455X (gfx1250) — compile-verified
//
#include <hip/hip_runtime.h>
#include <math.h>

typedef float v2f __attribute__((ext_vector_type(2)));
typedef float v8f __attribute__((ext_vector_type(8)));

constexpr int kNQ     = 64;     // sites
constexpr int kD      = 32;     // DCUT
constexpr int kHL     = 2;
constexpr int kNB     = 4096;   // batch
constexpr int kNT     = 32;     // batch tile per workgroup
constexpr int kM      = kHL * kD;       // 64 output rows
constexpr int kKSTEPS = 1088 / 4;       // 272 WMMA K-steps (1024 T + 32 Mh + 32 Mv)
constexpr int kAFSITE = kKSTEPS * 4 * 32;       // float2 per site (swizzled A)
constexpr int kGFLOATS = 64 * kD * kD;          // gamma scratch floats

// ---------------------------------------------------------------------------
// Prep 1: swizzle A_cat = [T | M_h | M_v] (64 x 1088 per site) into the exact
// per-lane layout of V_WMMA_F32_16X16X4_F32 A-fragments:
//   lane 0-15  : M = mt*16 + lane,      (x,y) = (K0,K1)
//   lane 16-31 : M = mt*16 + lane-16,   (x,y) = (K2,K3)
// AF index (float2): ((sid*272 + t)*4 + mt)*32 + lane
// ---------------------------------------------------------------------------
__global__ void prep_af_kernel(const float* __restrict__ T,
                               const float* __restrict__ Mh,
                               const float* __restrict__ Mv,
                               float* __restrict__ AF) {
  int gid = blockIdx.x * blockDim.x + threadIdx.x;
  if (gid >= 64 * kKSTEPS * 128) return;
  int lane = gid & 31;
  int mt   = (gid >> 5) & 3;
  int rest = gid >> 7;
  int t    = rest % kKSTEPS;
  int s    = rest / kKSTEPS;
  int m    = mt * 16 + (lane & 15);
  int o    = (lane < 16) ? 0 : 2;
  int k    = t * 4 + o;
  float x0, x1;
  if (k < 1024) {               // T site block: row m is contiguous 1024 floats
    const float* r = T + (size_t)s * 65536 + (size_t)m * 1024;
    x0 = r[k]; x1 = r[k + 1];
  } else if (k < 1056) {        // M_h block
    const float* r = Mh + (size_t)s * 2048 + (size_t)m * 32;
    x0 = r[k - 1024]; x1 = r[k - 1023];
  } else {                      // M_v block
    const float* r = Mv + (size_t)s * 2048 + (size_t)m * 32;
    x0 = r[k - 1056]; x1 = r[k - 1055];
  }
  AF[(size_t)gid * 2]     = x0;
  AF[(size_t)gid * 2 + 1] = x1;
}

// ---------------------------------------------------------------------------
// Prep 2: gamma[sid][i][j] = Re(eta @ eta^H)[i][j]
// ---------------------------------------------------------------------------
__global__ void prep_gamma_kernel(const float* __restrict__ eta,
                                  float* __restrict__ G) {
  int gid = blockIdx.x * blockDim.x + threadIdx.x;
  if (gid >= 64 * 1024) return;
  int s  = gid >> 10;
  int ij = gid & 1023;
  int ii = ij >> 5, jj = ij & 31;
  const float* e = eta + (size_t)s * 2048;    // (D, D, 2)
  float acc = 0.f;
  for (int k = 0; k < 32; ++k) {
    acc += e[(ii * 32 + k) * 2]     * e[(jj * 32 + k) * 2]
         + e[(ii * 32 + k) * 2 + 1] * e[(jj * 32 + k) * 2 + 1];
  }
  G[gid] = acc;
}

// ---------------------------------------------------------------------------
// Main: one workgroup per 32-batch tile; entire recurrent state in LDS.
// 8 waves: wave w -> output tile (mt = w&3, nt = w>>2) of the 64x32 GEMM.
// ---------------------------------------------------------------------------
__launch_bounds__(256)
__global__ void mpsrnn_main(const float* __restrict__ x,
                            const float* __restrict__ v,
                            const float* __restrict__ w_r,
                            const float* __restrict__ c_r,
                            const float* __restrict__ AF,
                            const float* __restrict__ G,
                            float* __restrict__ out) {
  __shared__ float ldsH[2 * 8 * kM * kNT];   // 2 snake rows x 8 cols of h (128KB)
  __shared__ float ldsHH[kD * kNT];          // gathered horizontal hidden (4KB)
  __shared__ float ldsHV[kD * kNT];          // gathered vertical hidden   (4KB)
  __shared__ float ldsG[kD * kD];            // gamma for current site     (4KB)
  __shared__ float ldsRed[8 * kNT];          // reductions                 (1KB)
  __shared__ float ldsInv[kNT];

  const int tid   = threadIdx.x;
  const int lane  = tid & 31;
  const int w     = tid >> 5;
  const int mt    = w & 3;
  const int ntile = w >> 2;
  const int nl    = lane & 15;
  const int nB    = ntile * 16 + nl;         // batch column for B/C/D fragments
  const int o     = (lane < 16) ? 0 : 2;     // K-pair select per lane half
  const int n0    = blockIdx.x * kNT;

  float amp = 1.f, phi = 0.f;                // live in wave 0, lane n

  for (int i = 0; i < kNQ; ++i) {
    const int b    = i >> 3;
    const int apos = i & 7;
    const int a    = (b & 1) ? (7 - apos) : apos;   // snake
    const int sid  = a * 8 + b;                     // parameter index [a][b]
    const int cur  = b & 1, prv = cur ^ 1;

    // neighbor selection (matches reference's h_store.get defaults)
    int hh_mode, hh_col = 0;                 // 0 zeros, 1 ones, 2 lds slot
    if ((a & 1) == 0) {
      if (a > 0 && (b & 1) == 0) { hh_mode = 2; hh_col = a - 1; }
      else hh_mode = 1;
    } else if (a == 7) {
      hh_mode = 0;
    } else {
      if (b & 1) { hh_mode = 2; hh_col = a + 1; }
      else hh_mode = 1;
    }
    const int hv_mode = (b == 0) ? 0 : 2;
    const int kprev   = (i > 0) ? (i - 1) : 0;

    // stage gamma for this site
    #pragma unroll
    for (int r = 0; r < 4; ++r) {
      int idx = tid + 256 * r;
      ldsG[idx] = G[sid * 1024 + idx];
    }
    // build gathered hh / hv tiles (HL-plane select per batch element)
    {
      const int n  = tid & 31;
      const int d0 = tid >> 5;
      const float xq = x[(n0 + n) * 64 + kprev];
      const int q = (xq > 0.f) ? 1 : 0;
      const float* srcH = &ldsH[((cur * 8 + hh_col) * kM + q * kD) * kNT];
      const float* srcV = &ldsH[((prv * 8 + a     ) * kM + q * kD) * kNT];
      #pragma unroll
      for (int r = 0; r < 4; ++r) {
        int d = d0 + 8 * r;
        float vh = (hh_mode == 2) ? srcH[d * kNT + n] : ((hh_mode == 1) ? 1.f : 0.f);
        float vv = (hv_mode == 2) ? srcV[d * kNT + n] : 0.f;
        ldsHH[d * kNT + n] = vh;
        ldsHV[d * kNT + n] = vv;
      }
    }
    __syncthreads();

    // ---- GEMM: H += A_cat x [hh (x) hv ; hh ; hv] via f32 WMMA ----
    v8f acc = {0.f, 0.f, 0.f, 0.f, 0.f, 0.f, 0.f, 0.f};
    const v2f* afp = ((const v2f*)AF) + (size_t)sid * kAFSITE + (mt * 32 + lane);
    for (int bi = 0; bi < 32; ++bi) {        // T region: K = bi*32 + c
      float hhb = ldsHH[bi * kNT + nB];
      #pragma unroll
      for (int cq = 0; cq < 8; ++cq) {
        v2f af = afp[(size_t)(bi * 8 + cq) * 128];
        int c = cq * 4 + o;
        v2f bf;
        bf.x = hhb * ldsHV[c * kNT + nB];
        bf.y = hhb * ldsHV[(c + 1) * kNT + nB];
        acc = __builtin_amdgcn_wmma_f32_16x16x4_f32(false, af, false, bf,
                                                    (short)0, acc, false, false);
      }
    }
    #pragma unroll
    for (int tt = 0; tt < 8; ++tt) {         // M_h region: B rows = hh
      v2f af = afp[(size_t)(256 + tt) * 128];
      int rr = tt * 4 + o;
      v2f bf;
      bf.x = ldsHH[rr * kNT + nB];
      bf.y = ldsHH[(rr + 1) * kNT + nB];
      acc = __builtin_amdgcn_wmma_f32_16x16x4_f32(false, af, false, bf,
                                                  (short)0, acc, false, false);
    }
    #pragma unroll
    for (int tt = 0; tt < 8; ++tt) {         // M_v region: B rows = hv
      v2f af = afp[(size_t)(264 + tt) * 128];
      int rr = tt * 4 + o;
      v2f bf;
      bf.x = ldsHV[rr * kNT + nB];
      bf.y = ldsHV[(rr + 1) * kNT + nB];
      acc = __builtin_amdgcn_wmma_f32_16x16x4_f32(false, af, false, bf,
                                                  (short)0, acc, false, false);
    }

    // bias add + store unnormalized h tile to this site's LDS slot
    float* hout = &ldsH[(cur * 8 + a) * kM * kNT];
    {
      const float* vs = v + sid * 64;
      const int mbase = mt * 16 + ((lane >> 4) << 3);   // C/D layout: VGPR j -> M=j / j+8
      #pragma unroll
      for (int j = 0; j < 8; ++j) {
        float val = acc[j] + vs[mbase + j];
        hout[(mbase + j) * kNT + nB] = val;
      }
    }
    __syncthreads();

    // ||h||^2 over all 64 rows, per batch column
    {
      const int n = tid & 31, g = tid >> 5;
      float s = 0.f;
      #pragma unroll
      for (int m = 0; m < 8; ++m) {
        float t_ = hout[(g * 8 + m) * kNT + n];
        s += t_ * t_;
      }
      ldsRed[g * kNT + n] = s;
    }
    __syncthreads();
    if (tid < 32) {
      float s = 1e-30f;
      #pragma unroll
      for (int g = 0; g < 8; ++g) s += ldsRed[g * kNT + tid];
      ldsInv[tid] = rsqrtf(s);
    }
    __syncthreads();
    {
      const int n = tid & 31;
      const float inv = ldsInv[n];
      #pragma unroll
      for (int r = 0; r < 8; ++r) {
        int m = (tid >> 5) + 8 * r;
        hout[m * kNT + n] *= inv;            // normalized h persists as state
      }
    }
    __syncthreads();

    // p[i_hl][n] = |h|^T gamma |h| — partials: wave w -> (i_hl = w&1, rows grp = w>>1)
    {
      const int i_hl = w & 1, grp = w >> 1;
      const int n = lane;
      const float* hq = &hout[i_hl * kD * kNT];
      float hav[32];
      #pragma unroll
      for (int d = 0; d < 32; ++d) hav[d] = fabsf(hq[d * kNT + n]);
      float ps = 0.f;
      #pragma unroll
      for (int aa = 0; aa < 8; ++aa) {
        const int arow = grp * 8 + aa;
        float inner = 0.f;
        #pragma unroll
        for (int bb = 0; bb < 32; ++bb) inner += ldsG[arow * 32 + bb] * hav[bb];
        ps += fabsf(hq[arow * kNT + n]) * inner;
      }
      ldsRed[w * kNT + n] = ps;
    }
    __syncthreads();

    // amplitude + phase update (wave 0, one lane per batch column)
    if (w == 0) {
      const int n = lane;
      float p0 = ldsRed[0 * kNT + n] + ldsRed[2 * kNT + n] + ldsRed[4 * kNT + n] + ldsRed[6 * kNT + n];
      float p1 = ldsRed[1 * kNT + n] + ldsRed[3 * kNT + n] + ldsRed[5 * kNT + n] + ldsRed[7 * kNT + n];
      const float xq = x[(n0 + n) * 64 + i];
      const int qi = (xq > 0.f) ? 1 : 0;
      float psel = (qi ? p1 : p0) / (p0 + p1 + 1e-30f);
      amp *= sqrtf(psel + 1e-30f);
      const float* hs = &hout[qi * kD * kNT];
      const float* wr = w_r + sid * 64;
      float zr = c_r[sid * 2], zi = c_r[sid * 2 + 1];
      #pragma unroll
      for (int d = 0; d < 32; ++d) {
        float h = hs[d * kNT + n];
        zr += wr[d * 2]     * h;
        zi += wr[d * 2 + 1] * h;
      }
      phi += atan2f(zi, zr);
    }
    __syncthreads();
  }

  if (w == 0) {
    const int n = lane;
    out[(size_t)(n0 + n) * 2]     = amp * cosf(phi);
    out[(size_t)(n0 + n) * 2 + 1] = amp * sinf(phi);
  }
}

// ---------------------------------------------------------------------------
extern "C" void kernel_launch(void* const* d_in, const int* in_sizes, int n_in,
                              void* d_out, int out_size, void* d_ws, size_t ws_size,
                              hipStream_t stream) {
  const float* x   = (const float*)d_in[0];
  const float* Mh  = (const float*)d_in[1];
  const float* Mv  = (const float*)d_in[2];
  const float* T   = (const float*)d_in[3];
  const float* v   = (const float*)d_in[4];
  const float* w_r = (const float*)d_in[5];
  const float* c_r = (const float*)d_in[6];
  const float* eta = (const float*)d_in[7];
  float* out = (float*)d_out;

  float* G  = (float*)d_ws;            // 64*1024 floats
  float* AF = G + kGFLOATS;            // 64*272*128*2 floats (~17.8MB)

  prep_gamma_kernel<<<(64 * 1024) / 256, 256, 0, stream>>>(eta, G);
  prep_af_kernel<<<(64 * kKSTEPS * 128) / 256, 256, 0, stream>>>(T, Mh, Mv, AF);
  mpsrnn_main<<<kNB / kNT, 256, 0, stream>>>(x, v, w_r, c_r, AF, G, out);
}